// RelRepBetweenContext_2834678415839
// MI455X (gfx1250) — compile-verified
//
#include <hip/hip_runtime.h>

// ---- problem constants (match reference) ----
#define B_   2
#define K_   16
#define S_   1024
#define H_   256
#define DIN  768      // 3*H
#define DFF  1152     // 1.5 * DIN
#define NPAIR (B_ * K_ * K_)   // 512

typedef __attribute__((ext_vector_type(2))) float v2f;
typedef __attribute__((ext_vector_type(8))) float v8f;

// Tied-operand WMMA: D is forced into the same 8-VGPR tuple as C, so the
// register allocator cannot rotate accumulators across the pipelined loop
// (eliminates the v_mov_b64 shuffles the SSA builtin form produced).
// Syntax per disasm of the builtin: v_wmma_f32_16x16x4_f32 D, A, B, C.
#define WMMA_ACC_F32(accv, av, bv)                                    \
    asm("v_wmma_f32_16x16x4_f32 %0, %1, %2, %0"                       \
        : "+v"(accv) : "v"(av), "v"(bv))

// ---------------------------------------------------------------------------
// Stage 1: build rel = [head | tail | between-context-maxpool]  (NPAIR x DIN)
// One block per (b,i,j) pair; thread = channel h. token_reps reads are
// coalesced across the 256 threads (H contiguous).
// ---------------------------------------------------------------------------
__global__ void build_rel_kernel(const float* __restrict__ csr,          // (B,K,H)
                                 const int*   __restrict__ ids,          // (B,K,2)
                                 const float* __restrict__ tok,          // (B,S,H)
                                 const unsigned char* __restrict__ tmask,// (B,S)
                                 const unsigned char* __restrict__ rmask,// (B,K*K)
                                 float* __restrict__ rel)                // (NPAIR,DIN)
{
    const int p  = blockIdx.x;           // pair index 0..NPAIR-1
    const int h  = threadIdx.x;          // channel 0..H-1
    const int b  = p / (K_ * K_);
    const int ij = p - b * K_ * K_;
    const int i  = ij / K_;
    const int j  = ij - i * K_;

    const int hs = ids[(b * K_ + i) * 2 + 0];
    const int he = ids[(b * K_ + i) * 2 + 1];
    const int ts = ids[(b * K_ + j) * 2 + 0];
    const int te = ids[(b * K_ + j) * 2 + 1];

    int lo = min(he, te);                // min_end
    int hi = max(hs, ts);                // max_start
    const bool valid = (lo < hi) && (rmask[b * K_ * K_ + ij] != 0);
    lo = max(lo, 0);
    hi = min(hi, S_);

    float m   = -__builtin_inff();
    bool  any = false;
    if (valid) {
        const float* tb = tok + ((size_t)b * S_) * H_ + h;
        for (int t = lo; t < hi; ++t) {
            if (tmask[b * S_ + t]) {
                m   = fmaxf(m, tb[(size_t)t * H_]);
                any = true;
            }
        }
    }
    const float ctx = any ? m : 0.0f;

    float* row = rel + (size_t)p * DIN;
    row[h]           = csr[((size_t)(b * K_ + i)) * H_ + h];   // head
    row[H_ + h]      = csr[((size_t)(b * K_ + j)) * H_ + h];   // tail
    row[2 * H_ + h]  = ctx;                                    // between-ctx
}

// ---------------------------------------------------------------------------
// Stage 2/3: f32 GEMM via V_WMMA_F32_16X16X4_F32, compile-time shapes.
// C(M,N) = A(M,KD) * Bm(KD,N) + bias, optional ReLU.
// One wave computes a 16(M) x 64(N) tile. Copy-free 2-deep software pipeline:
// two independent fragment buffers, each consumed by 4 WMMAs then refilled
// IN PLACE for k+8. Accumulators are pinned via tied-operand asm.
//
// ISA layouts (wave32, lane = half*16 + sub):
//   A 16x4 : VGPR v <- A[sub][2*half + v]   (one contiguous float2 per lane)
//   B 4x16 : VGPR v <- B[2*half + v][sub]   (coalesced across lanes)
//   D 16x16: VGPR r <- D[r + 8*half][sub]
// Requires M%16==0, N%64==0, KD%8==0 (512 / 1152,256 / 768,1152: all OK)
// ---------------------------------------------------------------------------
template <int M, int N, int KD, bool RELU>
__global__ void wmma_gemm_f32_kernel(const float* __restrict__ A,
                                     const float* __restrict__ Bm,
                                     const float* __restrict__ bias,
                                     float* __restrict__ C)
{
    constexpr int LDA = KD;
    constexpr int LDB = N;
    constexpr int LDC = N;
    constexpr int mTiles  = M / 16;
    constexpr int nSupers = N / 64;

    const int lane = threadIdx.x & 31;
    const int wave = threadIdx.x >> 5;
    const int w    = blockIdx.x * (blockDim.x >> 5) + wave;

    const int mt = w % mTiles;
    const int ns = w / mTiles;
    if (ns >= nSupers) return;            // wave-uniform exit (EXEC stays full)

    const int half = lane >> 4;           // 0 or 1
    const int sub  = lane & 15;
    const int kb   = half * 2;            // k-offset of this lane's fragment

    const float* ap = A + (size_t)(mt * 16 + sub) * LDA + kb;          // A[sub][kb]
    const float* bp = Bm + (size_t)kb * LDB + (ns * 64 + sub);         // B[kb][n]

    v8f acc0 = {}, acc1 = {}, acc2 = {}, acc3 = {};

    // ---- prologue: fill both buffers (k = 0 and k = 4) ----
    v2f a0 = *(const v2f*)ap;
    v2f a1 = *(const v2f*)(ap + 4);
    v2f b0[4], b1[4];
#pragma unroll
    for (int t = 0; t < 4; ++t) {
        b0[t].x = bp[0 * LDB + t * 16];   // B[kb  ][n+16t]
        b0[t].y = bp[1 * LDB + t * 16];   // B[kb+1][n+16t]
        b1[t].x = bp[4 * LDB + t * 16];
        b1[t].y = bp[5 * LDB + t * 16];
    }

    // ---- copy-free pipelined main loop (step 8) ----
    for (int k0 = 0; k0 < KD - 8; k0 += 8) {
        // consume buffer0 (k0), refill in place with k0+8
        WMMA_ACC_F32(acc0, a0, b0[0]);
        WMMA_ACC_F32(acc1, a0, b0[1]);
        WMMA_ACC_F32(acc2, a0, b0[2]);
        WMMA_ACC_F32(acc3, a0, b0[3]);
        a0 = *(const v2f*)(ap + 8);
#pragma unroll
        for (int t = 0; t < 4; ++t) {
            b0[t].x = bp[8 * LDB + t * 16];
            b0[t].y = bp[9 * LDB + t * 16];
        }
        // consume buffer1 (k0+4), refill in place with k0+12
        WMMA_ACC_F32(acc0, a1, b1[0]);
        WMMA_ACC_F32(acc1, a1, b1[1]);
        WMMA_ACC_F32(acc2, a1, b1[2]);
        WMMA_ACC_F32(acc3, a1, b1[3]);
        a1 = *(const v2f*)(ap + 12);
#pragma unroll
        for (int t = 0; t < 4; ++t) {
            b1[t].x = bp[12 * LDB + t * 16];
            b1[t].y = bp[13 * LDB + t * 16];
        }
        ap += 8;
        bp += (size_t)8 * LDB;
    }

    // ---- drain: buffers hold k = KD-8 and KD-4 ----
    WMMA_ACC_F32(acc0, a0, b0[0]);
    WMMA_ACC_F32(acc1, a0, b0[1]);
    WMMA_ACC_F32(acc2, a0, b0[2]);
    WMMA_ACC_F32(acc3, a0, b0[3]);
    WMMA_ACC_F32(acc0, a1, b1[0]);
    WMMA_ACC_F32(acc1, a1, b1[1]);
    WMMA_ACC_F32(acc2, a1, b1[2]);
    WMMA_ACC_F32(acc3, a1, b1[3]);

    // WMMA -> VALU read-after-write guard (worst case in ISA 7.12.1 is
    // 1 NOP + 8 co-exec slots; 8 wait-states covers it).
    asm volatile("s_nop 7");

    // ---- bias (+ReLU) epilogue, coalesced stores ----
    const v8f accs[4] = { acc0, acc1, acc2, acc3 };
#pragma unroll
    for (int t = 0; t < 4; ++t) {
        const int col = ns * 64 + t * 16 + sub;
        const float bv = bias[col];
#pragma unroll
        for (int r = 0; r < 8; ++r) {
            const int row = mt * 16 + r + half * 8;
            float v = accs[t][r] + bv;
            if (RELU) v = fmaxf(v, 0.0f);
            C[(size_t)row * LDC + col] = v;
        }
    }
}

// ---------------------------------------------------------------------------
extern "C" void kernel_launch(void* const* d_in, const int* in_sizes, int n_in,
                              void* d_out, int out_size, void* d_ws, size_t ws_size,
                              hipStream_t stream)
{
    const float*         csr   = (const float*)d_in[0];          // cand_span_reps (B,K,H)
    const int*           ids   = (const int*)d_in[1];            // cand_span_ids  (B,K,2)
    const float*         tok   = (const float*)d_in[2];          // token_reps     (B,S,H)
    const unsigned char* tmask = (const unsigned char*)d_in[3];  // token_masks    (B,S)
    const unsigned char* rmask = (const unsigned char*)d_in[4];  // rel_masks      (B,K*K)
    const float*         W1    = (const float*)d_in[5];          // (DIN,DFF)
    const float*         b1    = (const float*)d_in[6];          // (DFF)
    const float*         W2    = (const float*)d_in[7];          // (DFF,H)
    const float*         b2    = (const float*)d_in[8];          // (H)
    float*               out   = (float*)d_out;                  // (B,K*K,H)

    float* rel = (float*)d_ws;                       // NPAIR x DIN  (1.5 MB)
    float* hid = rel + (size_t)NPAIR * DIN;          // NPAIR x DFF  (2.3 MB)

    // Stage 1: build rel matrix (512 blocks x 256 threads = 8 waves/block)
    build_rel_kernel<<<NPAIR, H_, 0, stream>>>(csr, ids, tok, tmask, rmask, rel);

    // Stage 2: hid = relu(rel @ W1 + b1); M=512, N=1152, K=768
    {
        constexpr int waves  = (NPAIR / 16) * (DFF / 64);   // 32*18 = 576
        constexpr int blocks = waves / 8;                   // 72
        wmma_gemm_f32_kernel<NPAIR, DFF, DIN, true>
            <<<blocks, 256, 0, stream>>>(rel, W1, b1, hid);
    }

    // Stage 3: out = hid @ W2 + b2; M=512, N=256, K=1152
    {
        constexpr int waves  = (NPAIR / 16) * (H_ / 64);    // 32*4 = 128
        constexpr int blocks = waves / 8;                   // 16
        wmma_gemm_f32_kernel<NPAIR, H_, DFF, false>
            <<<blocks, 256, 0, stream>>>(hid, W2, b2, out);
    }
}